// S4DKernel_31911607009449
// MI455X (gfx1250) — compile-verified
//
#include <hip/hip_runtime.h>
#include <hip/hip_bf16.h>
#include <math.h>

// ---------------------------------------------------------------------------
// S4D (with the reference's elementwise-division "bug") collapses to a 2-tap
// causal filter per channel:
//   out[b,t,d] = (k0[d] + D[d]) * u[b,t,d] + k1[d] * u[b,t-1,d]
// Taps come from an 8-step repeated squaring of a per-channel 2x2 complex
// matrix. The streaming part is pure HBM bandwidth (~536MB @ 23.3TB/s), so we
// use the CDNA5 async-to-LDS path + nontemporal b128 stores, not WMMA.
// ---------------------------------------------------------------------------

#define D_MODEL 2048
#define SEQ_L   256
#define BATCHN  128
#define TILE_T  4                     // time rows computed per block
#define WIN_ROWS (TILE_T + 1)         // +1 previous row staged in LDS
#define ROW_F4  (D_MODEL / 4)         // 512 float4 per row

typedef float v4f __attribute__((ext_vector_type(4)));
typedef __attribute__((address_space(3))) float lds_f;

// ---------------- complex double helpers (setup kernel only) ---------------
struct C2 { double re, im; };
__device__ __forceinline__ C2 cmul(C2 a, C2 b) {
    return { a.re * b.re - a.im * b.im, a.re * b.im + a.im * b.re };
}
__device__ __forceinline__ C2 cadd(C2 a, C2 b) { return { a.re + b.re, a.im + b.im }; }
__device__ __forceinline__ C2 cdiv(C2 a, C2 b) {
    double d = b.re * b.re + b.im * b.im;
    return { (a.re * b.re + a.im * b.im) / d, (a.im * b.re - a.re * b.im) / d };
}

// ---------------------------------------------------------------------------
// Kernel 1: per-channel taps.  2048 threads total; cost is negligible.
// ---------------------------------------------------------------------------
__global__ __launch_bounds__(256)
void s4d_taps(const float* __restrict__ A_real, const float* __restrict__ A_imag,
              const float* __restrict__ B_real, const float* __restrict__ B_imag,
              const float* __restrict__ C_real, const float* __restrict__ C_imag,
              const float* __restrict__ Dvec,
              float* __restrict__ k0p, float* __restrict__ k1out)
{
    const int d = blockIdx.x * blockDim.x + threadIdx.x;
    if (d >= D_MODEL) return;

    C2 a[2], Bv[2], Cv[2];
#pragma unroll
    for (int j = 0; j < 2; ++j) {
        a[j]  = { -exp((double)A_real[d * 2 + j]), (double)A_imag[d * 2 + j] };
        Bv[j] = { (double)B_real[d * 2 + j], (double)B_imag[d * 2 + j] };
        Cv[j] = { (double)C_real[d * 2 + j], (double)C_imag[d * 2 + j] };
    }

    // dA[j][k] = (delta_jk + a_j/2) / (delta_jk - a_j/2)   (elementwise, DT=1)
    C2 M[2][2];
#pragma unroll
    for (int j = 0; j < 2; ++j)
#pragma unroll
        for (int k = 0; k < 2; ++k) {
            double del = (j == k) ? 1.0 : 0.0;
            C2 num = { del + 0.5 * a[j].re,  0.5 * a[j].im };
            C2 den = { del - 0.5 * a[j].re, -0.5 * a[j].im };
            M[j][k] = cdiv(num, den);
        }

    // M = dA^256 via 8 squarings (matches jnp.linalg.matrix_power binary expo)
#pragma unroll
    for (int s = 0; s < 8; ++s) {
        C2 t00 = cadd(cmul(M[0][0], M[0][0]), cmul(M[0][1], M[1][0]));
        C2 t01 = cadd(cmul(M[0][0], M[0][1]), cmul(M[0][1], M[1][1]));
        C2 t10 = cadd(cmul(M[1][0], M[0][0]), cmul(M[1][1], M[1][0]));
        C2 t11 = cadd(cmul(M[1][0], M[0][1]), cmul(M[1][1], M[1][1]));
        M[0][0] = t00; M[0][1] = t01; M[1][0] = t10; M[1][1] = t11;
    }

    // kernel[n] = Re( C_n * sum_h (I - M)[n][h] * B[h] )
    float kk[2];
#pragma unroll
    for (int n = 0; n < 2; ++n) {
        C2 acc = { 0.0, 0.0 };
#pragma unroll
        for (int h = 0; h < 2; ++h) {
            double del = (n == h) ? 1.0 : 0.0;
            C2 im = { del - M[n][h].re, -M[n][h].im };
            acc = cadd(acc, cmul(im, Bv[h]));
        }
        C2 r = cmul(Cv[n], acc);
        kk[n] = (float)r.re;
    }

    k0p[d]   = kk[0] + Dvec[d];   // fold skip connection into tap 0
    k1out[d] = kk[1];
}

// ---------------------------------------------------------------------------
// Kernel 2: bandwidth-bound streaming 2-tap filter.
// Each block: one batch b, TILE_T consecutive time rows, all 2048 channels.
// Stage WIN_ROWS x 2048 floats (40KB) into LDS with async-to-LDS b128 copies.
// ---------------------------------------------------------------------------
__global__ __launch_bounds__(256)
void s4d_stream(const float* __restrict__ u,
                const float* __restrict__ k0p,
                const float* __restrict__ k1,
                float* __restrict__ out)
{
    __shared__ float smem[WIN_ROWS * D_MODEL];   // 5 * 2048 * 4B = 40 KB

    const int tid    = threadIdx.x;
    const int chunks = SEQ_L / TILE_T;           // 64
    const int b      = blockIdx.x / chunks;
    const int t0     = (blockIdx.x % chunks) * TILE_T;

    const float* ubase = u + (size_t)b * SEQ_L * D_MODEL;
    lds_f* sbase = (lds_f*)smem;

    // Async-stage rows [t0-1 .. t0+TILE_T-1] (clamped at 0) into LDS.
    // 5 rows * 512 quads / 256 threads = 10 b128 async copies per thread.
#pragma unroll
    for (int i = 0; i < (WIN_ROWS * ROW_F4) / 256; ++i) {
        const int q   = i * 256 + tid;       // quad index in window
        const int row = q >> 9;              // / 512
        const int cq  = q & 511;             // quad within row
        int tr = t0 - 1 + row;
        if (tr < 0) tr = 0;                  // clamp; masked out in compute
        const float* gp = ubase + (size_t)tr * D_MODEL + cq * 4;
        lds_f* lp = sbase + (size_t)q * 4;
        asm volatile("global_load_async_to_lds_b128 %0, %1, off"
                     :: "v"(lp), "v"(gp) : "memory");
    }
    asm volatile("s_wait_asynccnt 0" ::: "memory");
    __syncthreads();

    // Per-thread taps for its two d-quads (d = tid*4 and 1024 + tid*4).
    const v4f* k0v = (const v4f*)k0p;
    const v4f* k1v = (const v4f*)k1;
    const v4f a0 = k0v[tid], a1 = k0v[256 + tid];
    const v4f b0 = k1v[tid], b1 = k1v[256 + tid];

    const v4f* s4   = (const v4f*)smem;
    float* obase    = out + (size_t)b * SEQ_L * D_MODEL;

#pragma unroll
    for (int r = 1; r < WIN_ROWS; ++r) {
        const int   t = t0 + r - 1;
        const float z = (t == 0) ? 0.0f : 1.0f;   // u[-1] == 0
        v4f* orow = (v4f*)(obase + (size_t)t * D_MODEL);
#pragma unroll
        for (int p = 0; p < 2; ++p) {
            const int q   = p * 256 + tid;
            const v4f cur = s4[r * ROW_F4 + q];
            const v4f prv = s4[(r - 1) * ROW_F4 + q];
            const v4f ka  = p ? a1 : a0;
            const v4f kb  = p ? b1 : b0;
            v4f o = ka * cur + (kb * prv) * z;
            __builtin_nontemporal_store(o, orow + q);   // output never re-read
        }
    }
}

// ---------------------------------------------------------------------------
extern "C" void kernel_launch(void* const* d_in, const int* in_sizes, int n_in,
                              void* d_out, int out_size, void* d_ws, size_t ws_size,
                              hipStream_t stream)
{
    const float* u      = (const float*)d_in[0];
    const float* A_real = (const float*)d_in[1];
    const float* A_imag = (const float*)d_in[2];
    const float* B_real = (const float*)d_in[3];
    const float* B_imag = (const float*)d_in[4];
    const float* C_real = (const float*)d_in[5];
    const float* C_imag = (const float*)d_in[6];
    const float* Dv     = (const float*)d_in[7];

    float* k0p = (float*)d_ws;         // 2048 floats: k0 + D
    float* k1  = k0p + D_MODEL;        // 2048 floats: k1

    s4d_taps<<<D_MODEL / 256, 256, 0, stream>>>(A_real, A_imag, B_real, B_imag,
                                                C_real, C_imag, Dv, k0p, k1);

    const int blocks = BATCHN * (SEQ_L / TILE_T);   // 8192
    s4d_stream<<<blocks, 256, 0, stream>>>(u, k0p, k1, (float*)d_out);
}